// BayesianFilter_83837761618422
// MI455X (gfx1250) — compile-verified
//
#include <hip/hip_runtime.h>
#include <math.h>
#include <stdint.h>

typedef __attribute__((ext_vector_type(2))) float v2f;
typedef __attribute__((ext_vector_type(8))) float v8f;

#define NSAMP   1024
#define NPTS    60
#define NB      1000
#define NBPAD   1008      // 63 tiles of 16
#define NTILES  63
#define BETA_SPEED 0.1f
#define BETA_CA    1.0f
#define BETA_BRAKE 1.0f
#define BETA_BND   1.0f
#define MAX_CA     19.6f
#define LOGCLIP   -73.682724f   // ln(1e-32)

// 16-byte async copy global -> LDS (gfx1250 GLOBAL_LOAD_ASYNC_TO_LDS_B128,
// GV addressing form, tracked by ASYNCcnt). lds_off is the 32-bit LDS byte
// offset (low 32 bits of the flat shared-aperture address).
__device__ __forceinline__ void async_copy16(uint32_t lds_off, const void* gptr) {
    uint64_t ga = (uint64_t)(uintptr_t)gptr;
    asm volatile("global_load_async_to_lds_b128 %0, %1, off"
                 :: "v"(lds_off), "v"(ga) : "memory");
}
__device__ __forceinline__ void wait_asynccnt0() {
    asm volatile("s_wait_asynccnt 0x0" ::: "memory");
}

// ---------------------------------------------------------------------------
// Kernel 1: one block (128 threads = 4 waves) per sample.
//   phase 1: async-stage padded boundary data into LDS (TDM-style async copy)
//   phase 2: threads 0..59 evaluate point/velocity/accel quantities
//   phase 3: all 4 waves run the WMMA nearest-boundary search (16 pts/wave)
//   phase 4: thread 0 reduces 60 points -> per-sample log-score
// ---------------------------------------------------------------------------
__global__ __launch_bounds__(128) void bf_sample_kernel(
    const float* __restrict__ curve,   // [8,2]
    const float* __restrict__ noise,   // [1024,8,2]
    const float* __restrict__ dT_p,    // [1]
    const float* __restrict__ sx,      // [20]
    const float* __restrict__ sy,      // [20]
    const float* __restrict__ M,       // [60,8]
    const float* __restrict__ Md,      // [60,7]
    const float* __restrict__ M2d,     // [60,6]
    const float* __restrict__ ibp, const float* __restrict__ ibn,   // [1000,2]
    const float* __restrict__ obp, const float* __restrict__ obn,   // [1000,2]
    float* __restrict__ logits)        // [1024]
{
    __shared__ __align__(16) float2 sh_bp[2 * NBPAD];   // boundary points (x,y)
    __shared__ __align__(16) float2 sh_bn[2 * NBPAD];   // boundary normals
    __shared__ float sh_px[64], sh_py[64];
    __shared__ float sh_spd[64], sh_bv[64], sh_ca[64], sh_ib[64], sh_ob[64];

    const int s   = blockIdx.x;
    const int tid = threadIdx.x;
    const float dT = dT_p[0];

    // ---- phase 1: boundary -> LDS via async copies
    // inner at [0,1008), outer at [1008,2016); 8000 B per array = 500 x 16 B
    {
        const uint32_t bp0 = (uint32_t)(uintptr_t)(&sh_bp[0]);
        const uint32_t bn0 = (uint32_t)(uintptr_t)(&sh_bn[0]);
        const uint32_t half = NBPAD * sizeof(float2);   // 8064, 16B aligned
        for (int i = tid; i < 500; i += 128) {
            const uint32_t off = (uint32_t)i * 16u;
            async_copy16(bp0 + off,        (const char*)ibp + off);
            async_copy16(bp0 + half + off, (const char*)obp + off);
            async_copy16(bn0 + off,        (const char*)ibn + off);
            async_copy16(bn0 + half + off, (const char*)obn + off);
        }
        // padding candidates: huge coords => never win the argmin
        if (tid < NBPAD - NB) {
            sh_bp[NB + tid]         = make_float2(1.0e18f, 1.0e18f);
            sh_bp[NBPAD + NB + tid] = make_float2(1.0e18f, 1.0e18f);
            sh_bn[NB + tid]         = make_float2(0.0f, 0.0f);
            sh_bn[NBPAD + NB + tid] = make_float2(0.0f, 0.0f);
        }
    }
    if (tid < 64) {
        sh_px[tid] = 0.0f; sh_py[tid] = 0.0f;
        sh_spd[tid] = 0.0f; sh_bv[tid] = 0.0f; sh_ca[tid] = 0.0f;
        sh_ib[tid] = -1.0e30f; sh_ob[tid] = -1.0e30f;
    }

    // ---- phase 2: per-point geometry (threads 0..59)
    if (tid < NPTS) {
        float Px[8], Py[8];
#pragma unroll
        for (int k = 0; k < 8; ++k) {
            Px[k] = curve[2 * k]     + noise[s * 16 + 2 * k];
            Py[k] = curve[2 * k + 1] + noise[s * 16 + 2 * k + 1];
        }
        // position
        float posx = 0.0f, posy = 0.0f;
#pragma unroll
        for (int k = 0; k < 8; ++k) {
            float w = M[tid * 8 + k];
            posx += w * Px[k]; posy += w * Py[k];
        }
        // first derivative
        float vx = 0.0f, vy = 0.0f;
#pragma unroll
        for (int k = 0; k < 7; ++k) {
            float w = Md[tid * 7 + k];
            vx += w * (Px[k + 1] - Px[k]);
            vy += w * (Py[k + 1] - Py[k]);
        }
        vx *= 7.0f / dT;  vy *= 7.0f / dT;
        float spd = sqrtf(vx * vx + vy * vy);
        float utx = vx / spd, uty = vy / spd;
        // second derivative
        float ax = 0.0f, ay = 0.0f;
#pragma unroll
        for (int k = 0; k < 6; ++k) {
            float w = M2d[tid * 6 + k];
            ax += w * ((Px[k + 2] - Px[k + 1]) - (Px[k + 1] - Px[k]));
            ay += w * ((Py[k + 2] - Py[k + 1]) - (Py[k + 1] - Py[k]));
        }
        float inv_dT2 = 1.0f / (dT * dT);
        ax *= 42.0f * inv_dT2;  ay *= 42.0f * inv_dT2;

        float lin = ax * utx + ay * uty;
        float cx = ax - utx * lin, cy = ay - uty * lin;
        float ca = sqrtf(cx * cx + cy * cy);

        // piecewise-linear braking-limit lookup (sx is uniform linspace)
        float x0 = sx[0];
        float h  = sx[1] - x0;
        float u  = (spd - x0) / h;
        int   ii = (int)floorf(u);
        ii = ii < 0 ? 0 : (ii > 18 ? 18 : ii);
        float fr = u - (float)ii;
        fr = fr < 0.0f ? 0.0f : (fr > 1.0f ? 1.0f : fr);
        float lim = sy[ii] + fr * (sy[ii + 1] - sy[ii]);

        float bv = fminf(lin - lim, 0.0f);
        bv = fmaxf(bv, -1.0e9f);

        sh_px[tid] = posx; sh_py[tid] = posy;
        sh_spd[tid] = spd; sh_bv[tid] = bv; sh_ca[tid] = ca;
    }
    wait_asynccnt0();     // this wave's async copies landed in LDS
    __syncthreads();      // all waves' copies + point data visible

    // ---- phase 3: WMMA nearest-boundary search -------------------------------
    // score(c, p) = |b_c|^2 - 2 b_c . p    (rank-equivalent to |p - b_c|^2)
    // A (16x4 f32): row = candidate c: [bx, by, b2, 0]
    //   lane m<16  supplies K0,K1 = (bx, by); lane m+16 supplies K2,K3 = (b2, 0)
    // B (4x16 f32): col = point n: [-2px, -2py, 1, 0]
    //   lane n<16  supplies K0,K1 = (-2px, -2py); lane n+16 supplies K2,K3 = (1, 0)
    // C: lane l holds column N=l&15, rows M = r + (l>=16 ? 8 : 0), r = vgpr idx
    const int  lane = tid & 31;
    const int  wv   = tid >> 5;
    const int  nn   = lane & 15;
    const int  pt   = wv * 16 + nn;   // point this lane-column represents
    const bool hi   = lane >= 16;

    const float px = sh_px[pt], py = sh_py[pt];
    v2f Bop;
    Bop.x = hi ? 1.0f : (-2.0f * px);
    Bop.y = hi ? 0.0f : (-2.0f * py);

#pragma unroll 1
    for (int bnd = 0; bnd < 2; ++bnd) {
        const int base = bnd * NBPAD;
        float minv = 3.0e38f;
        int   mini = base;
#pragma unroll 1
        for (int t = 0; t < NTILES; ++t) {
            const int c = base + t * 16 + nn;      // candidate row for this lane
            const float2 b = sh_bp[c];             // one ds_load_b64
            const float b2 = fmaf(b.x, b.x, b.y * b.y);
            v2f Aop;
            Aop.x = hi ? b2   : b.x;
            Aop.y = hi ? 0.0f : b.y;
            v8f Cz = {0.0f, 0.0f, 0.0f, 0.0f, 0.0f, 0.0f, 0.0f, 0.0f};
            v8f S = __builtin_amdgcn_wmma_f32_16x16x4_f32(
                        false, Aop, false, Bop, (short)0, Cz, false, false);
            const int idx0 = base + t * 16 + (hi ? 8 : 0);
#pragma unroll
            for (int r = 0; r < 8; ++r) {
                const float v = S[r];
                if (v < minv) { minv = v; mini = idx0 + r; }
            }
        }
        // merge the two row-halves (lane <-> lane^16); prefer lower index on tie
        const float ov = __shfl_xor(minv, 16, 32);
        const int   oi = __shfl_xor(mini, 16, 32);
        if (ov < minv || (ov == minv && oi < mini)) { minv = ov; mini = oi; }

        if (!hi && pt < NPTS) {
            const float2 cb = sh_bp[mini];
            const float2 cn = sh_bn[mini];
            const float d = (cb.x - px) * cn.x + (cb.y - py) * cn.y;
            if (bnd == 0) sh_ib[pt] = d; else sh_ob[pt] = d;
        }
    }
    __syncthreads();

    // ---- phase 4: per-sample reduction -> log-score
    if (tid == 0) {
        float ssum = 0.0f, bmin = 0.0f, camax = -1.0e30f;
        float ibmax = -1.0e30f, obmax = -1.0e30f;
        for (int p = 0; p < NPTS; ++p) {
            ssum  += sh_spd[p];
            bmin   = fminf(bmin, sh_bv[p]);
            camax  = fmaxf(camax, sh_ca[p]);
            ibmax  = fmaxf(ibmax, sh_ib[p]);
            obmax  = fmaxf(obmax, sh_ob[p]);
        }
        const float avg  = ssum / (float)NPTS;
        const float ca_d = fmaxf(camax - MAX_CA, 0.0f);
        const float om   = fmaxf(fmaxf(ibmax, 0.0f), fmaxf(obmax, 0.0f));
        const float l = BETA_SPEED * avg
                      + BETA_BRAKE * bmin
                      - BETA_CA * ca_d
                      + fmaxf(-BETA_BND * om, LOGCLIP);
        logits[s] = l;
    }
}

// ---------------------------------------------------------------------------
// Kernel 2: softmax over 1024 samples + weighted average of control points.
// One block of 1024 threads (32 waves), deterministic tree reductions.
// ---------------------------------------------------------------------------
__global__ __launch_bounds__(1024) void bf_reduce_kernel(
    const float* __restrict__ curve,   // [8,2]
    const float* __restrict__ noise,   // [1024,8,2]
    const float* __restrict__ logits,  // [1024]
    float* __restrict__ out)           // [8,2]
{
    __shared__ float sh_m[32];
    __shared__ float sh_acc[32][17];

    const int tid  = threadIdx.x;
    const int lane = tid & 31;
    const int wv   = tid >> 5;

    __builtin_prefetch(noise + tid * 16, 0, 0);   // global_prefetch_b8

    const float l = logits[tid];

    // block-wide max of logits
    float m = l;
#pragma unroll
    for (int d = 16; d > 0; d >>= 1) m = fmaxf(m, __shfl_xor(m, d, 32));
    if (lane == 0) sh_m[wv] = m;
    __syncthreads();
    if (tid == 0) {
        float mm = sh_m[0];
        for (int i = 1; i < 32; ++i) mm = fmaxf(mm, sh_m[i]);
        sh_m[0] = mm;
    }
    __syncthreads();
    m = sh_m[0];

    const float w = expf(l - m);

    float vals[17];
#pragma unroll
    for (int j = 0; j < 16; ++j) vals[j] = w * noise[tid * 16 + j];
    vals[16] = w;

#pragma unroll
    for (int j = 0; j < 17; ++j) {
        float v = vals[j];
#pragma unroll
        for (int d = 16; d > 0; d >>= 1) v += __shfl_xor(v, d, 32);
        vals[j] = v;
    }
    if (lane == 0) {
#pragma unroll
        for (int j = 0; j < 17; ++j) sh_acc[wv][j] = vals[j];
    }
    __syncthreads();

    if (tid < 17) {
        float acc = 0.0f;
        for (int i = 0; i < 32; ++i) acc += sh_acc[i][tid];
        sh_acc[0][tid] = acc;
    }
    __syncthreads();

    if (tid < 16) {
        out[tid] = curve[tid] + sh_acc[0][tid] / sh_acc[0][16];
    }
}

// ---------------------------------------------------------------------------
extern "C" void kernel_launch(void* const* d_in, const int* in_sizes, int n_in,
                              void* d_out, int out_size, void* d_ws, size_t ws_size,
                              hipStream_t stream) {
    (void)in_sizes; (void)n_in; (void)out_size; (void)ws_size;
    const float* curve = (const float*)d_in[0];
    const float* noise = (const float*)d_in[1];
    const float* dT    = (const float*)d_in[2];
    const float* sx    = (const float*)d_in[3];
    const float* sy    = (const float*)d_in[4];
    const float* M     = (const float*)d_in[5];
    const float* Md    = (const float*)d_in[6];
    const float* M2d   = (const float*)d_in[7];
    const float* ibp   = (const float*)d_in[8];
    const float* ibn   = (const float*)d_in[9];
    const float* obp   = (const float*)d_in[10];
    const float* obn   = (const float*)d_in[11];

    float* logits = (float*)d_ws;

    bf_sample_kernel<<<NSAMP, 128, 0, stream>>>(curve, noise, dT, sx, sy,
                                                M, Md, M2d,
                                                ibp, ibn, obp, obn, logits);
    bf_reduce_kernel<<<1, 1024, 0, stream>>>(curve, noise, logits, (float*)d_out);
}